// GAT_40235253629330
// MI455X (gfx1250) — compile-verified
//
#include <hip/hip_runtime.h>

#define N_NODES 50000
#define IN_CH   128
#define HID     32
#define HEADS   4
#define HC      128     // HEADS*HID
#define OUT_CH  16
#define BATCH   512
#define NEG_SLOPE 0.2f

typedef float v2f __attribute__((ext_vector_type(2)));
typedef float v8f __attribute__((ext_vector_type(8)));

// ---------------------------------------------------------------------------
// f32 WMMA 16x16 tile layouts (wave32):
//   A (16x4): lanes 0-15 -> M=0..15, K={0,1}; lanes 16-31 -> K={2,3}.
//   B (4x16): lanes 0-15 -> N=0..15, K={0,1}; lanes 16-31 -> K={2,3}.
//   C/D:      VGPR r: lanes 0-15 -> M=r, lanes 16-31 -> M=r+8, N=lane&15.
// ---------------------------------------------------------------------------

// K1: h = x @ W1.  25000 tiles (3125 m-tiles x 8 n-tiles), one wave each.
__global__ __launch_bounds__(128) void gat_gemm_h(const float* __restrict__ x,
                                                  const float* __restrict__ W1,
                                                  float* __restrict__ h) {
    const int tile = blockIdx.x * 4 + (threadIdx.x >> 5);
    const int m0 = (tile >> 3) * 16;
    const int n0 = (tile & 7) * 16;
    const int lane = threadIdx.x & 31;
    const int half = lane >> 4;
    const int l    = lane & 15;
    const float* __restrict__ xrow = x + (size_t)(m0 + l) * IN_CH;

    v8f acc = {0.f, 0.f, 0.f, 0.f, 0.f, 0.f, 0.f, 0.f};
    for (int k0 = 0; k0 < IN_CH; k0 += 4) {
        const int ka = k0 + half * 2;
        v2f a, b;
        a.x = xrow[ka];
        a.y = xrow[ka + 1];
        b.x = W1[(size_t)ka * HC + n0 + l];
        b.y = W1[(size_t)(ka + 1) * HC + n0 + l];
        acc = __builtin_amdgcn_wmma_f32_16x16x4_f32(
            false, a, false, b, (short)0, acc, false, false);
    }
#pragma unroll
    for (int r = 0; r < 8; ++r) {
        const int row = m0 + r + half * 8;
        h[(size_t)row * HC + n0 + l] = acc[r];
    }
}

// K2: per-(node,head) attention coefficients (h is L2-resident).
__global__ void gat_attn_coef(const float* __restrict__ h,
                              const float* __restrict__ att_src,
                              const float* __restrict__ att_dst,
                              float* __restrict__ a_s, float* __restrict__ a_d) {
    const int i = blockIdx.x * blockDim.x + threadIdx.x;   // n*HEADS + head
    if (i >= N_NODES * HEADS) return;
    const int head = i & (HEADS - 1);
    const float* __restrict__ hp = h + (size_t)(i >> 2) * HC + head * HID;
    float s = 0.f, d = 0.f;
#pragma unroll
    for (int c = 0; c < HID; ++c) {
        s += hp[c] * att_src[head * HID + c];
        d += hp[c] * att_dst[head * HID + c];
    }
    a_s[i] = s;
    a_d[i] = d;
}

// K3: zero scratch accumulators
__global__ void gat_init_ws(float* __restrict__ agg, float* __restrict__ den,
                            unsigned* __restrict__ amax, float* __restrict__ g) {
    const int i = blockIdx.x * blockDim.x + threadIdx.x;
    if (i < N_NODES * HC) agg[i] = 0.f;
    if (i < N_NODES * HEADS) { den[i] = 0.f; amax[i] = 0u; }
    if (i < BATCH * HC) g[i] = 0.f;
}

// -------- helpers ----------------------------------------------------------
__device__ __forceinline__ void edge_sd(const int* __restrict__ ei, int E, int e,
                                        int& s, int& d) {
    if (e < E) { s = ei[e]; d = ei[E + e]; }
    else       { s = e - E; d = e - E; }          // self loops
}
__device__ __forceinline__ float lrelu(float x) { return x > 0.f ? x : NEG_SLOPE * x; }
// order-preserving float <-> uint for atomic segment-max
__device__ __forceinline__ unsigned ordf(float f) {
    unsigned u = __float_as_uint(f);
    return (u & 0x80000000u) ? ~u : (u | 0x80000000u);
}
__device__ __forceinline__ float unordf(unsigned u) {
    return (u & 0x80000000u) ? __uint_as_float(u & 0x7fffffffu) : __uint_as_float(~u);
}

// K4: per-edge segment max of leaky-relu logits
__global__ void gat_edge_max(const int* __restrict__ ei, int E,
                             const float* __restrict__ a_s,
                             const float* __restrict__ a_d,
                             unsigned* __restrict__ amax) {
    const int e = blockIdx.x * blockDim.x + threadIdx.x;
    if (e >= E + N_NODES) return;
    int s, d; edge_sd(ei, E, e, s, d);
#pragma unroll
    for (int hh = 0; hh < HEADS; ++hh) {
        const float al = lrelu(a_s[s * HEADS + hh] + a_d[d * HEADS + hh]);
        atomicMax(&amax[d * HEADS + hh], ordf(al));
    }
}

// K5: per-edge softmax denominator
__global__ void gat_edge_den(const int* __restrict__ ei, int E,
                             const float* __restrict__ a_s,
                             const float* __restrict__ a_d,
                             const unsigned* __restrict__ amax,
                             float* __restrict__ den) {
    const int e = blockIdx.x * blockDim.x + threadIdx.x;
    if (e >= E + N_NODES) return;
    int s, d; edge_sd(ei, E, e, s, d);
#pragma unroll
    for (int hh = 0; hh < HEADS; ++hh) {
        const float al = lrelu(a_s[s * HEADS + hh] + a_d[d * HEADS + hh]);
        const float ex = __expf(al - unordf(amax[d * HEADS + hh]));
        atomicAdd(&den[d * HEADS + hh], ex);
    }
}

// K6: per-edge weighted scatter agg[dst] += a * h[src].
// One wave32 per edge; lane = channel within head (HID==32). Edge index is
// wave-uniform -> readfirstlane so the softmax inputs take the scalar path;
// only the 128B gather + 128B atomic burst per head stays vector.
__global__ __launch_bounds__(256) void gat_edge_agg(const int* __restrict__ ei, int E,
                                                    const float* __restrict__ a_s,
                                                    const float* __restrict__ a_d,
                                                    const unsigned* __restrict__ amax,
                                                    const float* __restrict__ den,
                                                    const float* __restrict__ h,
                                                    float* __restrict__ agg) {
    const int e = blockIdx.x * (blockDim.x >> 5) + (threadIdx.x >> 5);
    if (e >= E + N_NODES) return;            // wave-uniform exit
    const int lane = threadIdx.x & 31;
    int s, d;
    if (e < E) {                              // wave-uniform branch
        s = __builtin_amdgcn_readfirstlane(ei[e]);
        d = __builtin_amdgcn_readfirstlane(ei[E + e]);
    } else {
        s = e - E; d = s;
    }
    float aw[HEADS];
#pragma unroll
    for (int hh = 0; hh < HEADS; ++hh) {
        const float al = lrelu(a_s[s * HEADS + hh] + a_d[d * HEADS + hh]);
        aw[hh] = __expf(al - unordf(amax[d * HEADS + hh]))
                 / (den[d * HEADS + hh] + 1e-16f);
    }
#pragma unroll
    for (int hh = 0; hh < HEADS; ++hh) {
        const int j = hh * HID + lane;
        atomicAdd(&agg[(size_t)d * HC + j], h[(size_t)s * HC + j] * aw[hh]);
    }
}

// K7: bias + ELU + global_add_pool into g[BATCH, HC]
__global__ void gat_node_pool(const float* __restrict__ agg,
                              const float* __restrict__ b1,
                              const int* __restrict__ batch,
                              float* __restrict__ g) {
    const int i = blockIdx.x * blockDim.x + threadIdx.x;
    if (i >= N_NODES * HC) return;
    const int n = i >> 7, j = i & 127;
    float v = agg[i] + b1[j];
    v = v > 0.f ? v : (__expf(v) - 1.f);
    atomicAdd(&g[(size_t)batch[n] * HC + j], v);
}

// K8: g1 = g @ lin1_w + lin1_b  (512x128x32) — 64 WMMA tiles, one wave each
__global__ __launch_bounds__(128) void gat_mlp1_wmma(const float* __restrict__ g,
                                                     const float* __restrict__ w,
                                                     const float* __restrict__ b,
                                                     float* __restrict__ g1) {
    const int tile = blockIdx.x * 4 + (threadIdx.x >> 5);  // 32 m-tiles x 2 n-tiles
    const int m0 = (tile >> 1) * 16;
    const int n0 = (tile & 1) * 16;
    const int lane = threadIdx.x & 31;
    const int half = lane >> 4, l = lane & 15;
    const float* __restrict__ grow = g + (size_t)(m0 + l) * HC;
    v8f acc = {0.f, 0.f, 0.f, 0.f, 0.f, 0.f, 0.f, 0.f};
    for (int k0 = 0; k0 < HC; k0 += 4) {
        const int ka = k0 + half * 2;
        v2f a, bb;
        a.x = grow[ka];
        a.y = grow[ka + 1];
        bb.x = w[ka * HID + n0 + l];
        bb.y = w[(ka + 1) * HID + n0 + l];
        acc = __builtin_amdgcn_wmma_f32_16x16x4_f32(
            false, a, false, bb, (short)0, acc, false, false);
    }
    const float bias = b[n0 + l];
#pragma unroll
    for (int r = 0; r < 8; ++r) {
        const int row = m0 + r + half * 8;
        g1[row * HID + n0 + l] = acc[r] + bias;
    }
}

// K9: out = g1 @ lin2_w + lin2_b  (512x32x16) — 32 WMMA tiles, one wave each
__global__ __launch_bounds__(128) void gat_mlp2_wmma(const float* __restrict__ g1,
                                                     const float* __restrict__ w,
                                                     const float* __restrict__ b,
                                                     float* __restrict__ out) {
    const int tile = blockIdx.x * 4 + (threadIdx.x >> 5);  // 32 m-tiles, 1 n-tile
    const int m0 = tile * 16;
    const int lane = threadIdx.x & 31;
    const int half = lane >> 4, l = lane & 15;
    const float* __restrict__ grow = g1 + (size_t)(m0 + l) * HID;
    v8f acc = {0.f, 0.f, 0.f, 0.f, 0.f, 0.f, 0.f, 0.f};
    for (int k0 = 0; k0 < HID; k0 += 4) {
        const int ka = k0 + half * 2;
        v2f a, bb;
        a.x = grow[ka];
        a.y = grow[ka + 1];
        bb.x = w[ka * OUT_CH + l];
        bb.y = w[(ka + 1) * OUT_CH + l];
        acc = __builtin_amdgcn_wmma_f32_16x16x4_f32(
            false, a, false, bb, (short)0, acc, false, false);
    }
    const float bias = b[l];
#pragma unroll
    for (int r = 0; r < 8; ++r) {
        out[(m0 + r + half * 8) * OUT_CH + l] = acc[r] + bias;
    }
}

// ---------------------------------------------------------------------------
extern "C" void kernel_launch(void* const* d_in, const int* in_sizes, int n_in,
                              void* d_out, int out_size, void* d_ws, size_t ws_size,
                              hipStream_t stream) {
    const float* x        = (const float*)d_in[0];
    const int*   ei       = (const int*)d_in[1];
    const int*   batch    = (const int*)d_in[2];
    const float* W1       = (const float*)d_in[3];
    const float* att_src  = (const float*)d_in[4];
    const float* att_dst  = (const float*)d_in[5];
    const float* b1       = (const float*)d_in[6];
    const float* lin1_w   = (const float*)d_in[7];
    const float* lin1_b   = (const float*)d_in[8];
    const float* lin2_w   = (const float*)d_in[9];
    const float* lin2_b   = (const float*)d_in[10];
    float* out = (float*)d_out;

    const int E    = in_sizes[1] / 2;
    const int Etot = E + N_NODES;

    // workspace carve-up (~55 MB)
    char* ws = (char*)d_ws;
    float*    h    = (float*)ws;    ws += (size_t)N_NODES * HC * sizeof(float);
    float*    agg  = (float*)ws;    ws += (size_t)N_NODES * HC * sizeof(float);
    float*    a_s  = (float*)ws;    ws += (size_t)N_NODES * HEADS * sizeof(float);
    float*    a_d  = (float*)ws;    ws += (size_t)N_NODES * HEADS * sizeof(float);
    unsigned* amax = (unsigned*)ws; ws += (size_t)N_NODES * HEADS * sizeof(unsigned);
    float*    den  = (float*)ws;    ws += (size_t)N_NODES * HEADS * sizeof(float);
    float*    g    = (float*)ws;    ws += (size_t)BATCH * HC * sizeof(float);
    float*    g1   = (float*)ws;

    const int tiles = (N_NODES / 16) * (HC / 16);   // 25000, divisible by 4
    gat_gemm_h<<<tiles / 4, 128, 0, stream>>>(x, W1, h);
    gat_attn_coef<<<(N_NODES * HEADS + 255) / 256, 256, 0, stream>>>(h, att_src, att_dst, a_s, a_d);
    gat_init_ws<<<(N_NODES * HC + 255) / 256, 256, 0, stream>>>(agg, den, amax, g);
    gat_edge_max<<<(Etot + 255) / 256, 256, 0, stream>>>(ei, E, a_s, a_d, amax);
    gat_edge_den<<<(Etot + 255) / 256, 256, 0, stream>>>(ei, E, a_s, a_d, amax, den);
    gat_edge_agg<<<(Etot + 7) / 8, 256, 0, stream>>>(ei, E, a_s, a_d, amax, den, h, agg);
    gat_node_pool<<<(N_NODES * HC + 255) / 256, 256, 0, stream>>>(agg, b1, batch, g);
    gat_mlp1_wmma<<<16, 128, 0, stream>>>(g, lin1_w, lin1_b, g1);   // 64 tiles
    gat_mlp2_wmma<<<8, 128, 0, stream>>>(g1, lin2_w, lin2_b, out);  // 32 tiles
}